// SwarmBrain_48833778155896
// MI455X (gfx1250) — compile-verified
//
#include <hip/hip_runtime.h>
#include <math.h>

#define THREADS 256

typedef __attribute__((ext_vector_type(2))) float v2f;
typedef __attribute__((ext_vector_type(8))) float v8f;

static __device__ __forceinline__ void atomicAddF32(float* p, float v) {
  __hip_atomic_fetch_add(p, v, __ATOMIC_RELAXED, __HIP_MEMORY_SCOPE_AGENT);
}

// ---------------- utility ----------------
__global__ __launch_bounds__(THREADS) void k_zero(float* __restrict__ p, int n) {
  int t = blockIdx.x * blockDim.x + threadIdx.x;
  if (t < n) p[t] = 0.0f;
}

// deg[dst[e]] += 1  (float; exact for counts << 2^24)
__global__ __launch_bounds__(THREADS) void k_deg(const int* __restrict__ dst,
                                                 float* __restrict__ deg, int e) {
  int t = blockIdx.x * blockDim.x + threadIdx.x;
  if (t < e) atomicAddF32(&deg[dst[t]], 1.0f);
}

// in-place deg -> deg^{-1/2} (0 where deg==0)
__global__ __launch_bounds__(THREADS) void k_dinv(float* __restrict__ deg, int n) {
  int t = blockIdx.x * blockDim.x + threadIdx.x;
  if (t < n) {
    float d = deg[t];
    deg[t] = (d > 0.0f) ? (1.0f / sqrtf(d)) : 0.0f;
  }
}

// ---------------- layer-1 GEMM: [N,5] @ [5,32] (tiny, scalar) ----------------
__global__ __launch_bounds__(THREADS) void k_gemm5(const float* __restrict__ x,
                                                   const float* __restrict__ W,
                                                   float* __restrict__ xw, int n) {
  int t = blockIdx.x * blockDim.x + threadIdx.x;
  if (t >= n * 32) return;
  int row = t >> 5, col = t & 31;
  float s = 0.0f;
#pragma unroll
  for (int k = 0; k < 5; ++k) s += x[row * 5 + k] * W[k * 32 + col];
  xw[t] = s;
}

// ---------------- 32x32 GEMM via V_WMMA_F32_16X16X4_F32 ----------------
// One wave computes a 16x32 output strip: rows [wave*16, wave*16+16), all 32 cols.
// A(16x4 f32): lane holds row (lane&15); half-wave selects K pair {0,1} vs {2,3}.
// B(4x16 f32): mirrored K split; N striped across lanes within VGPR.
// C/D(16x16 f32): 8 VGPRs; VGPR r -> M = r (+8 for lanes 16..31), N = lane&15.
__global__ __launch_bounds__(THREADS) void k_gemm32_wmma(const float* __restrict__ Hin,
                                                         const float* __restrict__ W, // [32,32]
                                                         float* __restrict__ Xw,
                                                         int nTiles) {
  int wave = blockIdx.x * (blockDim.x >> 5) + (threadIdx.x >> 5);
  if (wave >= nTiles) return;
  int lane   = threadIdx.x & 31;
  int rowSel = lane & 15;
  int hi     = lane >> 4;            // 0: lanes 0-15, 1: lanes 16-31
  int row    = wave * 16 + rowSel;

  v8f c0 = {};  // cols 0..15
  v8f c1 = {};  // cols 16..31
#pragma unroll
  for (int ks = 0; ks < 8; ++ks) {
    int k0 = ks * 4 + hi * 2;
    v2f a;
    a.x = Hin[row * 32 + k0];
    a.y = Hin[row * 32 + k0 + 1];
    v2f b0, b1;
    b0.x = W[k0 * 32 + rowSel];
    b0.y = W[(k0 + 1) * 32 + rowSel];
    b1.x = W[k0 * 32 + 16 + rowSel];
    b1.y = W[(k0 + 1) * 32 + 16 + rowSel];
    c0 = __builtin_amdgcn_wmma_f32_16x16x4_f32(false, a, false, b0, (short)0, c0, false, false);
    c1 = __builtin_amdgcn_wmma_f32_16x16x4_f32(false, a, false, b1, (short)0, c1, false, false);
  }
  int outRow = wave * 16 + hi * 8;
#pragma unroll
  for (int r = 0; r < 8; ++r) {
    Xw[(outRow + r) * 32 + rowSel]      = c0[r];
    Xw[(outRow + r) * 32 + 16 + rowSel] = c1[r];
  }
}

// ---------------- edge message passing (L2-bandwidth bound) ----------------
// lane = feature; wave covers one full 128B row gather + 128B of scatter atomics
__global__ __launch_bounds__(THREADS) void k_edge(const int* __restrict__ src,
                                                  const int* __restrict__ dst,
                                                  const float* __restrict__ dinv,
                                                  const float* __restrict__ xw,
                                                  float* __restrict__ agg, int e) {
  int t = blockIdx.x * blockDim.x + threadIdx.x;
  if (t >= e * 32) return;
  int ed = t >> 5, f = t & 31;
  int s = src[ed], d = dst[ed];
  float nm = dinv[s] * dinv[d];
  atomicAddF32(&agg[d * 32 + f], nm * xw[s * 32 + f]);
}

// h = relu(h + b)  in place
__global__ __launch_bounds__(THREADS) void k_bias_relu(float* __restrict__ h,
                                                       const float* __restrict__ b, int n) {
  int t = blockIdx.x * blockDim.x + threadIdx.x;
  if (t < n) h[t] = fmaxf(h[t] + b[t & 31], 0.0f);
}

// ---------------- heads: distance & chase scores ----------------
__global__ __launch_bounds__(THREADS) void k_heads(const float* __restrict__ h,
                                                   const float* __restrict__ Wd,
                                                   const float* __restrict__ bd,
                                                   const float* __restrict__ Wc,
                                                   const float* __restrict__ bc,
                                                   float* __restrict__ out, int n) {
  int i = blockIdx.x * blockDim.x + threadIdx.x;
  if (i >= n) return;
  const float* hr = h + (size_t)i * 32;
  float sd = 0.0f, sc = 0.0f;
#pragma unroll
  for (int k = 0; k < 32; ++k) {
    float hv = hr[k];
    sd += hv * Wd[k];
    sc += hv * Wc[k];
  }
  out[i]     = sd + bd[0];
  out[n + i] = sc + bc[0];
}

// argmax over chase scores; first occurrence on ties (matches jnp.argmax)
__global__ __launch_bounds__(256) void k_argmax(const float* __restrict__ s, int n,
                                                int* __restrict__ outIdx) {
  __shared__ float sv[256];
  __shared__ int   si[256];
  int t = threadIdx.x;
  float bv = -__builtin_huge_valf();
  int bi = n;
  for (int j = t; j < n; j += 256) {
    float v = s[j];
    if (v > bv) { bv = v; bi = j; }  // j ascending per thread -> keeps first
  }
  sv[t] = bv; si[t] = bi;
  __syncthreads();
  for (int off = 128; off > 0; off >>= 1) {
    if (t < off) {
      if (sv[t + off] > sv[t] || (sv[t + off] == sv[t] && si[t + off] < si[t])) {
        sv[t] = sv[t + off];
        si[t] = si[t + off];
      }
    }
    __syncthreads();
  }
  if (t == 0) *outIdx = si[0];
}

// target row -> [2] location score + [9] action logits
__global__ __launch_bounds__(32) void k_target(const float* __restrict__ h,
                                               const int* __restrict__ tidx,
                                               const float* __restrict__ Wt,
                                               const float* __restrict__ bt,
                                               const float* __restrict__ Wa,
                                               const float* __restrict__ ba,
                                               float* __restrict__ out, int n) {
  int j = threadIdx.x;
  int ti = *tidx;
  const float* ht = h + (size_t)ti * 32;
  if (j < 2) {
    float s = bt[j];
#pragma unroll
    for (int k = 0; k < 32; ++k) s += ht[k] * Wt[k * 2 + j];
    out[2 * n + j] = s;
  } else if (j < 11) {
    int a = j - 2;
    float s = ba[a];
#pragma unroll
    for (int k = 0; k < 32; ++k) s += ht[k] * Wa[k * 9 + a];
    out[2 * n + 2 + a] = s;
  }
}

extern "C" void kernel_launch(void* const* d_in, const int* in_sizes, int n_in,
                              void* d_out, int out_size, void* d_ws, size_t ws_size,
                              hipStream_t stream) {
  const float* x  = (const float*)d_in[0];
  const int*   ei = (const int*)d_in[1];
  const float *W1 = (const float*)d_in[2],  *b1 = (const float*)d_in[3];
  const float *W2 = (const float*)d_in[4],  *b2 = (const float*)d_in[5];
  const float *W3 = (const float*)d_in[6],  *b3 = (const float*)d_in[7];
  const float *Wd = (const float*)d_in[8],  *bd = (const float*)d_in[9];
  const float *Wc = (const float*)d_in[10], *bc = (const float*)d_in[11];
  const float *Wt = (const float*)d_in[12], *bt = (const float*)d_in[13];
  const float *Wa = (const float*)d_in[14], *ba = (const float*)d_in[15];

  const int N = in_sizes[0] / 5;   // 50000
  const int E = in_sizes[1] / 2;   // 1600000
  const int* src = ei;
  const int* dst = ei + E;

  // workspace: dinv[N] | xw[N*32] | h[N*32] | tidx (int)  (~13 MB)
  float* wsf  = (float*)d_ws;
  float* dinv = wsf;
  float* xw   = dinv + N;
  float* h    = xw + (size_t)N * 32;
  int*   tidx = (int*)(h + (size_t)N * 32);

  float* out = (float*)d_out;

  const int NF = N * 32;
  const int EF = E * 32;
  const int nTiles = N / 16;  // 3125 (exact)
  auto nb = [](int n) { return (n + THREADS - 1) / THREADS; };
  const int gemmBlocks = (nTiles + (THREADS / 32) - 1) / (THREADS / 32);

  // degree -> dinv (edge_index invariant across layers: compute once)
  k_zero<<<nb(N), THREADS, 0, stream>>>(dinv, N);
  k_deg<<<nb(E), THREADS, 0, stream>>>(dst, dinv, E);
  k_dinv<<<nb(N), THREADS, 0, stream>>>(dinv, N);

  // ---- layer 1 ----
  k_gemm5<<<nb(NF), THREADS, 0, stream>>>(x, W1, xw, N);
  k_zero<<<nb(NF), THREADS, 0, stream>>>(h, NF);
  k_edge<<<nb(EF), THREADS, 0, stream>>>(src, dst, dinv, xw, h, E);
  k_bias_relu<<<nb(NF), THREADS, 0, stream>>>(h, b1, NF);

  // ---- layer 2 ----
  k_gemm32_wmma<<<gemmBlocks, THREADS, 0, stream>>>(h, W2, xw, nTiles);
  k_zero<<<nb(NF), THREADS, 0, stream>>>(h, NF);
  k_edge<<<nb(EF), THREADS, 0, stream>>>(src, dst, dinv, xw, h, E);
  k_bias_relu<<<nb(NF), THREADS, 0, stream>>>(h, b2, NF);

  // ---- layer 3 ----
  k_gemm32_wmma<<<gemmBlocks, THREADS, 0, stream>>>(h, W3, xw, nTiles);
  k_zero<<<nb(NF), THREADS, 0, stream>>>(h, NF);
  k_edge<<<nb(EF), THREADS, 0, stream>>>(src, dst, dinv, xw, h, E);
  k_bias_relu<<<nb(NF), THREADS, 0, stream>>>(h, b3, NF);

  // ---- heads ----
  k_heads<<<nb(N), THREADS, 0, stream>>>(h, Wd, bd, Wc, bc, out, N);
  k_argmax<<<1, 256, 0, stream>>>(out + N, N, tidx);
  k_target<<<1, 32, 0, stream>>>(h, tidx, Wt, bt, Wa, ba, out, N);
}